// HausdorffERLoss_59227599012640
// MI455X (gfx1250) — compile-verified
//
#include <hip/hip_runtime.h>

// ---------------------------------------------------------------------------
// HausdorffER loss: 10 iterations of 5-point cross stencil + ReLU(x-0.5) +
// per-sample min/max normalization + weighted accumulate; final scalar mean.
// Memory-bound (~2.6 GB of traffic => ~110us floor @ 23.3 TB/s).
// CDNA5 path: async global->LDS tile staging (ASYNCcnt) + s_wait_asynccnt.
// Normalization is folded algebraically into the next conv (conv is linear),
// so each iteration is exactly one read + one write of the 128 MB field.
// ---------------------------------------------------------------------------

static constexpr int IMG_H   = 1024;
static constexpr int IMG_W   = 1024;
static constexpr int BATCH   = 32;
static constexpr int NPIX    = IMG_H * IMG_W;      // per sample
static constexpr int TILE_H  = 16;
static constexpr int TILE_W  = 256;
static constexpr int LDSW    = 264;                // LDS row stride (floats)
static constexpr int XOFF    = 4;                  // interior col offset (16B aligned)
static constexpr int NEROS   = 10;
static constexpr int BLK_PER_IMG = (IMG_W / TILE_W) * (IMG_H / TILE_H); // 256

#if __has_builtin(__builtin_amdgcn_global_load_async_to_lds_b128) && \
    __has_builtin(__builtin_amdgcn_global_load_async_to_lds_b32)
#define USE_ASYNC_LDS 1
#else
#define USE_ASYNC_LDS 0
#endif

typedef int v4i_gcc __attribute__((vector_size(16)));   // matches builtin param

__device__ __forceinline__ void lds_copy_b128(float* lds_dst, const float* gsrc) {
#if USE_ASYNC_LDS
  __builtin_amdgcn_global_load_async_to_lds_b128(
      (__attribute__((address_space(1))) v4i_gcc*)(const void*)gsrc,
      (__attribute__((address_space(3))) v4i_gcc*)(void*)lds_dst, 0, 0);
#else
  *(float4*)lds_dst = *(const float4*)gsrc;
#endif
}

__device__ __forceinline__ void lds_copy_b32(float* lds_dst, const float* gsrc) {
#if USE_ASYNC_LDS
  __builtin_amdgcn_global_load_async_to_lds_b32(
      (__attribute__((address_space(1))) int*)(const void*)gsrc,
      (__attribute__((address_space(3))) int*)(void*)lds_dst, 0, 0);
#else
  *lds_dst = *gsrc;
#endif
}

__device__ __forceinline__ void wait_async_lds() {
#if USE_ASYNC_LDS
#if __has_builtin(__builtin_amdgcn_s_wait_asynccnt)
  __builtin_amdgcn_s_wait_asynccnt(0);
#else
  asm volatile("s_wait_asynccnt 0" ::: "memory");
#endif
#endif
}

// One erosion step on one (TILE_H x TILE_W) tile of one sample.
// FUSED=true : input field is (pred-target)^2 computed on the fly (iter 0).
// FUSED=false: input is raw ero from previous iter; normalization of the
//              previous field is folded in: dil = s*conv5(raw) - s*m*w(x,y).
template <bool FUSED>
__global__ __launch_bounds__(256) void erode_step(
    const float* __restrict__ in0,   // FUSED: pred, else raw field
    const float* __restrict__ in1,   // FUSED: target, else unused
    float* __restrict__ out,         // raw ero for this iteration
    const float* __restrict__ scl,   // per-sample 1/ptp of prev iter
    const float* __restrict__ moff,  // per-sample min of prev iter
    unsigned* __restrict__ g_min,    // per-sample running min (float bits)
    unsigned* __restrict__ g_max,    // per-sample running max (float bits)
    float* __restrict__ psum)        // per-block partial sums [BATCH][256]
{
  __shared__ float tile[18 * LDSW];
  __shared__ float red[256];

  const int tx = threadIdx.x;            // 0..63
  const int ty = threadIdx.y;            // 0..3
  const int t  = ty * 64 + tx;           // 0..255
  const int x0 = blockIdx.x * TILE_W;
  const int y0 = blockIdx.y * TILE_H;
  const int b  = blockIdx.z;
  const size_t sbase = (size_t)b * NPIX;

  const float* p0 = in0 + sbase;
  const float* p1 = FUSED ? (in1 + sbase) : (const float*)nullptr;

  // ---- stage tile rows y0-1 .. y0+16, interior cols, as b128 chunks -------
  for (int chunk = t; chunk < 18 * 64; chunk += 256) {
    const int r  = chunk >> 6;           // lds row 0..17
    const int cc = (chunk & 63) << 2;    // col within tile, multiple of 4
    const int gy = y0 - 1 + r;
    float* ldst = &tile[r * LDSW + XOFF + cc];
    if (gy >= 0 && gy < IMG_H) {
      const size_t gi = (size_t)gy * IMG_W + (x0 + cc);
      if (FUSED) {
        float4 a = *(const float4*)(p0 + gi);
        float4 c = *(const float4*)(p1 + gi);
        float4 v;
        v.x = (a.x - c.x) * (a.x - c.x);
        v.y = (a.y - c.y) * (a.y - c.y);
        v.z = (a.z - c.z) * (a.z - c.z);
        v.w = (a.w - c.w) * (a.w - c.w);
        *(float4*)ldst = v;
      } else {
        lds_copy_b128(ldst, p0 + gi);
      }
    } else {
      *(float4*)ldst = make_float4(0.f, 0.f, 0.f, 0.f);  // zero pad
    }
  }
  // ---- halo columns (x0-1 and x0+TILE_W), 18 rows x 2 --------------------
  if (t < 36) {
    const int r    = t >> 1;
    const int side = t & 1;
    const int gy = y0 - 1 + r;
    const int gx = side ? (x0 + TILE_W) : (x0 - 1);
    float* ldst = &tile[r * LDSW + (side ? (XOFF + TILE_W) : (XOFF - 1))];
    if (gy >= 0 && gy < IMG_H && gx >= 0 && gx < IMG_W) {
      const size_t gi = (size_t)gy * IMG_W + gx;
      if (FUSED) {
        float a = p0[gi], c = p1[gi];
        *ldst = (a - c) * (a - c);
      } else {
        lds_copy_b32(ldst, p0 + gi);
      }
    } else {
      *ldst = 0.0f;
    }
  }
  wait_async_lds();     // each wave drains its own ASYNCcnt before the barrier
  __syncthreads();

  // ---- stencil: 4 rows x 4 contiguous cols per thread ---------------------
  float s, sm;
  if (FUSED) { s = 1.0f; sm = 0.0f; }
  else       { s = scl[b]; sm = s * moff[b]; }

  float* outp = out + sbase;
  float lmin = __uint_as_float(0x7f800000u);  // +inf
  float lmax = 0.0f;                          // values are >= 0
  float lsum = 0.0f;
  const int cbase = tx << 2;
  const int gxb   = x0 + cbase;

#pragma unroll
  for (int jr = 0; jr < 4; ++jr) {
    const int r  = ty + (jr << 2);       // output row within tile, 0..15
    const int gy = y0 + r;
    const float yb = (float)((gy == 0) + (gy == IMG_H - 1));
    const float* rowU = &tile[(r    ) * LDSW + XOFF + cbase];
    const float* rowC = &tile[(r + 1) * LDSW + XOFF + cbase];
    const float* rowD = &tile[(r + 2) * LDSW + XOFF + cbase];
    float res[4];
#pragma unroll
    for (int i = 0; i < 4; ++i) {
      const int gx = gxb + i;
      const float conv5 = (rowU[i] + rowD[i] + rowC[i - 1] + rowC[i + 1] + rowC[i]) * 0.2f;
      // # in-image taps / 5  (zero padding correction for folded normalization)
      const float wcnt = 5.0f - yb - (float)((gx == 0) + (gx == IMG_W - 1));
      const float dil  = s * conv5 - sm * (wcnt * 0.2f);
      const float e    = fmaxf(dil - 0.5f, 0.0f);
      res[i] = e;
      lsum += e;
      lmin = fminf(lmin, e);
      lmax = fmaxf(lmax, e);
    }
    float4 o; o.x = res[0]; o.y = res[1]; o.z = res[2]; o.w = res[3];
    *(float4*)(outp + (size_t)gy * IMG_W + gxb) = o;
  }

  // ---- block reductions ---------------------------------------------------
  red[t] = lsum; __syncthreads();
  for (int sft = 128; sft; sft >>= 1) { if (t < sft) red[t] += red[t + sft]; __syncthreads(); }
  if (t == 0)
    psum[((size_t)b << 8) + blockIdx.y * gridDim.x + blockIdx.x] = red[0];
  __syncthreads();

  red[t] = lmin; __syncthreads();
  for (int sft = 128; sft; sft >>= 1) { if (t < sft) red[t] = fminf(red[t], red[t + sft]); __syncthreads(); }
  if (t == 0) atomicMin(&g_min[b], __float_as_uint(red[0]));   // bits monotone: x >= 0
  __syncthreads();

  red[t] = lmax; __syncthreads();
  for (int sft = 128; sft; sft >>= 1) { if (t < sft) red[t] = fmaxf(red[t], red[t + sft]); __syncthreads(); }
  if (t == 0) atomicMax(&g_max[b], __float_as_uint(red[0]));
}

// Per-iteration bookkeeping: one wave, one lane per sample. Deterministic
// serial reduce of the 256 per-block partial sums; produces next iteration's
// (scale, offset), resets min/max, accumulates the loss scalar.
__global__ void erode_stats(unsigned* __restrict__ g_min, unsigned* __restrict__ g_max,
                            const float* __restrict__ psum,
                            float* __restrict__ scl, float* __restrict__ moff,
                            float* __restrict__ lacc,
                            float wk, int last, float* __restrict__ out)
{
  __shared__ float contrib[BATCH];
  const int b = threadIdx.x;
  const float mn = __uint_as_float(g_min[b]);
  const float mx = __uint_as_float(g_max[b]);
  float sum = 0.0f;
  const float* pp = psum + ((size_t)b << 8);
  for (int i = 0; i < BLK_PER_IMG; ++i) sum += pp[i];

  const float ptp = mx - mn;
  float c, s, m;
  if (ptp != 0.0f) {
    s = 1.0f / ptp; m = mn;
    c = (sum - (float)NPIX * mn) * s;    // sum of normalized field
  } else {
    s = 1.0f; m = 0.0f; c = sum;         // reference keeps raw field
  }
  scl[b] = s; moff[b] = m;
  g_min[b] = 0x7f800000u;                // reset for next iteration
  g_max[b] = 0u;
  contrib[b] = c;
  __syncthreads();
  if (b == 0) {
    float tot = 0.0f;
    for (int i = 0; i < BATCH; ++i) tot += contrib[i];
    const float acc = *lacc + wk * tot;
    *lacc = acc;
    if (last) *out = acc * (1.0f / ((float)BATCH * (float)NPIX));
  }
}

__global__ void erode_init(unsigned* g_min, unsigned* g_max, float* lacc) {
  const int i = threadIdx.x;
  if (i < BATCH) { g_min[i] = 0x7f800000u; g_max[i] = 0u; }
  if (i == 0) *lacc = 0.0f;
}

extern "C" void kernel_launch(void* const* d_in, const int* in_sizes, int n_in,
                              void* d_out, int out_size, void* d_ws, size_t ws_size,
                              hipStream_t stream) {
  (void)in_sizes; (void)n_in; (void)out_size; (void)ws_size;
  const float* pred   = (const float*)d_in[0];
  const float* target = (const float*)d_in[1];
  float* out = (float*)d_out;

  char* ws = (char*)d_ws;
  const size_t NB = (size_t)BATCH * NPIX * sizeof(float);   // 128 MB
  float* bufA = (float*)ws;
  float* bufB = (float*)(ws + NB);
  float*    smallf = (float*)(ws + 2 * NB);
  unsigned* g_min = (unsigned*)smallf;          // 32
  unsigned* g_max = (unsigned*)(smallf + 32);   // 32
  float*    scl   = smallf + 64;                // 32
  float*    moff  = smallf + 96;                // 32
  float*    lacc  = smallf + 128;               // 1
  float*    psum  = smallf + 160;               // 32*256

  dim3 block(64, 4, 1);
  dim3 grid(IMG_W / TILE_W, IMG_H / TILE_H, BATCH);

  erode_init<<<1, 64, 0, stream>>>(g_min, g_max, lacc);

  // k = 0: bound=(pred-target)^2 fused into the first stencil pass
  erode_step<true><<<grid, block, 0, stream>>>(pred, target, bufA,
                                               nullptr, nullptr,
                                               g_min, g_max, psum);
  erode_stats<<<1, BATCH, 0, stream>>>(g_min, g_max, psum, scl, moff, lacc,
                                       1.0f, 0, out);

  float* src = bufA;
  float* dst = bufB;
  for (int k = 1; k < NEROS; ++k) {
    erode_step<false><<<grid, block, 0, stream>>>(src, nullptr, dst,
                                                  scl, moff,
                                                  g_min, g_max, psum);
    const float wk = (float)((k + 1) * (k + 1));   // (k+1)^ALPHA, ALPHA=2
    erode_stats<<<1, BATCH, 0, stream>>>(g_min, g_max, psum, scl, moff, lacc,
                                         wk, (k == NEROS - 1) ? 1 : 0, out);
    float* tmp = src; src = dst; dst = tmp;
  }
}